// HungarianMatcherNLoss_49873160241421
// MI455X (gfx1250) — compile-verified
//
#include <hip/hip_runtime.h>
#include <hip/hip_bf16.h>
#include <math.h>

// Problem constants from the reference
#define BB 2
#define KK 256
#define DESC 256
#define BK (BB * KK)      // 512 flat keypoints
#define MAXN BK           // max rows after keep-filter
#define FEAT_STRIDE (2 + DESC)  // 258

typedef __attribute__((ext_vector_type(2))) float v2f;
typedef __attribute__((ext_vector_type(8))) float v8f;

#if defined(__has_builtin)
#if __has_builtin(__builtin_amdgcn_global_load_async_to_lds_b64)
#define HAVE_ASYNC_LDS 1
#endif
#endif

#ifdef HAVE_ASYNC_LDS
// b64 async copy moves one 2xi32 vector per lane: (int2 AS1*, int2 AS3*, imm, imm)
typedef int async_i2 __attribute__((ext_vector_type(2)));
#define AS1_V2I(p) ((__attribute__((address_space(1))) async_i2*)(p))
#define AS3_V2I(p) ((__attribute__((address_space(3))) async_i2*)(p))
static __device__ __forceinline__ void wait_asynccnt0() {
#if __has_builtin(__builtin_amdgcn_s_wait_asynccnt)
    __builtin_amdgcn_s_wait_asynccnt(0);
#else
    asm volatile("s_wait_asynccnt 0" ::: "memory");
#endif
}
#endif

// ---------------------------------------------------------------------------
// Kernel 1: keypoint transform via V_WMMA_F32_16X16X4_F32
//   cal = round(M(4x4) @ all_kp(4xN)); o_tf = (cal_x + w/2, -cal_y + h/2)
// One wave handles 16 points per WMMA. A = M padded to 16x4, B = 4x16 points.
// ---------------------------------------------------------------------------
__global__ void transform_kernel(const float* __restrict__ theta,
                                 const float* __restrict__ feat,
                                 const int* __restrict__ wptr,
                                 const int* __restrict__ hptr,
                                 float* __restrict__ flat_cal) {
    const int lane = threadIdx.x & 31;
    const int wave = threadIdx.x >> 5;      // 8 waves (256 threads)
    const float w = (float)wptr[0];
    const float h = (float)hptr[0];
    const int m  = lane & 15;               // row of A / column of B
    const int kh = lane >> 4;               // K half (0: K=0,1   1: K=2,3)

    for (int chunk = wave; chunk < BK / 16; chunk += 8) {
        const int batch = (chunk * 16) / KK;    // chunks never straddle batches
        const float th = theta[batch];
        const float c = cosf(th), s = sinf(th);

        // A row m: [c,-s,0,0] / [s,c,0,0] / [0,0,1,0] / [0,0,0,1] / zeros
        const float A0 = (m == 0) ? c : ((m == 1) ? s : 0.0f);
        const float A1 = (m == 0) ? -s : ((m == 1) ? c : 0.0f);
        const float A2 = (m == 2) ? 1.0f : 0.0f;
        const float A3 = (m == 3) ? 1.0f : 0.0f;
        v2f a;
        a.x = kh ? A2 : A0;   // A[m][2*kh + 0]
        a.y = kh ? A3 : A1;   // A[m][2*kh + 1]

        // B column n = point (chunk*16 + n): [x', y', 0, 1]^T
        const int pt = chunk * 16 + m;      // lanes 16-31 mirror lanes 0-15
        const float kx = feat[pt * FEAT_STRIDE + 0];
        const float ky = feat[pt * FEAT_STRIDE + 1];
        const float xp = kx - 0.5f * w;
        const float yp = -ky + 0.5f * h;
        v2f b;
        b.x = kh ? 0.0f : xp;  // B[2*kh + 0][n]
        b.y = kh ? 1.0f : yp;  // B[2*kh + 1][n]

        v8f cacc = {0.f, 0.f, 0.f, 0.f, 0.f, 0.f, 0.f, 0.f};
        // D = A x B + 0   (EXEC is all-ones here)
        v8f d = __builtin_amdgcn_wmma_f32_16x16x4_f32(
            false, a, false, b, (short)0, cacc, false, false);

        // D row0 = cal_x, row1 = cal_y for points N=0..15 (lanes 0..15)
        if (lane < 16) {
            const float cal_x = rintf(d[0]);   // jnp.round == RNE
            const float cal_y = rintf(d[1]);
            flat_cal[pt * 2 + 0] = cal_x + 0.5f * w;
            flat_cal[pt * 2 + 1] = -cal_y + 0.5f * h;
        }
    }
}

// ---------------------------------------------------------------------------
// Kernel 2: keep-filter + stream compaction (order preserving), 512 threads
// ---------------------------------------------------------------------------
__global__ void compact_kernel(const float* __restrict__ tf_feat,
                               const float* __restrict__ flat_cal,
                               int* __restrict__ n_keep,
                               int* __restrict__ keep_idx,
                               float* __restrict__ f_pred,
                               float* __restrict__ f_cal) {
    __shared__ int scan[BK];
    const int t = threadIdx.x;
    const float ox = flat_cal[t * 2 + 0];
    const float oy = flat_cal[t * 2 + 1];
    const int flag = (ox >= 0.0f && oy >= 0.0f) ? 1 : 0;
    scan[t] = flag;
    __syncthreads();
    for (int off = 1; off < BK; off <<= 1) {   // inclusive Hillis-Steele scan
        int v = (t >= off) ? scan[t - off] : 0;
        __syncthreads();
        scan[t] += v;
        __syncthreads();
    }
    const int pos = scan[t] - flag;            // exclusive
    if (flag) {
        keep_idx[pos] = t;
        f_pred[pos * 2 + 0] = tf_feat[t * FEAT_STRIDE + 0];
        f_pred[pos * 2 + 1] = tf_feat[t * FEAT_STRIDE + 1];
        f_cal[pos * 2 + 0] = ox;
        f_cal[pos * 2 + 1] = oy;
    }
    if (t == BK - 1) *n_keep = scan[BK - 1];
}

// ---------------------------------------------------------------------------
// Kernel 3: L1 cdist tiles (16x16 per block).
// Descriptor tiles staged into LDS with CDNA5 async global->LDS copies
// (GLOBAL_LOAD_ASYNC_TO_LDS_B64, ASYNCcnt) when the toolchain exposes them.
// Descriptor rows start at byte offset row*1032 + 8, so even-d elements are
// 8-byte aligned -> b64 copies are legal.
// ---------------------------------------------------------------------------
__global__ void cdist_kernel(const float* __restrict__ feat,
                             const float* __restrict__ tf_feat,
                             const int* __restrict__ n_keep_p,
                             const int* __restrict__ keep_idx,
                             const float* __restrict__ f_pred,
                             const float* __restrict__ f_cal,
                             float* __restrict__ cost_kp,
                             float* __restrict__ cost_desc) {
    const int n = *n_keep_p;
    const int i0 = blockIdx.y * 16;
    const int j0 = blockIdx.x * 16;
    if (i0 >= n || j0 >= n) return;

    __shared__ float At[16][DESC];
    __shared__ float Bt[16][DESC];
    const int tx = threadIdx.x, ty = threadIdx.y;
    const int tid = ty * 16 + tx;

    __builtin_prefetch(&f_pred[i0 * 2], 0, 3);   // global_prefetch_b8
    __builtin_prefetch(&f_cal[j0 * 2], 0, 3);

#ifdef HAVE_ASYNC_LDS
    // 16 rows x 128 b64 copies per side; each wave waits on its own ASYNCcnt.
    for (int e = tid; e < 16 * (DESC / 2); e += 256) {
        const int r = e >> 7;              // e / 128
        const int d2 = (e & 127) * 2;      // even element index
        const int ia = i0 + r;
        const int rowA = keep_idx[(ia < n) ? ia : (n - 1)];
        __builtin_amdgcn_global_load_async_to_lds_b64(
            AS1_V2I(feat + rowA * FEAT_STRIDE + 2 + d2),
            AS3_V2I(&At[r][d2]), 0, 0);
        const int jb = j0 + r;
        const int rowB = keep_idx[(jb < n) ? jb : (n - 1)];
        __builtin_amdgcn_global_load_async_to_lds_b64(
            AS1_V2I(tf_feat + rowB * FEAT_STRIDE + 2 + d2),
            AS3_V2I(&Bt[r][d2]), 0, 0);
    }
    wait_asynccnt0();        // this wave's async copies are in LDS
    __syncthreads();         // make all waves' copies visible
#else
    for (int e = tid; e < 16 * DESC; e += 256) {
        const int r = e >> 8;
        const int d = e & (DESC - 1);
        const int ia = i0 + r;
        const int rowA = keep_idx[(ia < n) ? ia : (n - 1)];
        At[r][d] = feat[rowA * FEAT_STRIDE + 2 + d];
        const int jb = j0 + r;
        const int rowB = keep_idx[(jb < n) ? jb : (n - 1)];
        Bt[r][d] = tf_feat[rowB * FEAT_STRIDE + 2 + d];
    }
    __syncthreads();
#endif

    const int i = i0 + ty, j = j0 + tx;
    if (i < n && j < n) {
        float sd = 0.0f;
#pragma unroll 8
        for (int d = 0; d < DESC; ++d) sd += fabsf(At[ty][d] - Bt[tx][d]);
        cost_desc[i * n + j] = sd;
        const float ck = fabsf(f_pred[i * 2 + 0] - f_cal[j * 2 + 0]) +
                         fabsf(f_pred[i * 2 + 1] - f_cal[j * 2 + 1]);
        cost_kp[i * n + j] = ck;
    }
}

// ---------------------------------------------------------------------------
// Kernel 4: deterministic sums of both matrices -> ratio
// ---------------------------------------------------------------------------
__global__ void reduce_kernel(const int* __restrict__ n_keep_p,
                              const float* __restrict__ cost_kp,
                              const float* __restrict__ cost_desc,
                              float* __restrict__ sums) {
    __shared__ float s1[1024];
    __shared__ float s2[1024];
    const int n = *n_keep_p;
    const int total = n * n;
    const int t = threadIdx.x;
    float a = 0.0f, b = 0.0f;
    for (int idx = t; idx < total; idx += 1024) {
        a += cost_kp[idx];
        b += cost_desc[idx];
    }
    s1[t] = a; s2[t] = b;
    __syncthreads();
    for (int s = 512; s > 0; s >>= 1) {
        if (t < s) { s1[t] += s1[t + s]; s2[t] += s2[t + s]; }
        __syncthreads();
    }
    if (t == 0) {
        sums[0] = s1[0];
        sums[1] = s2[0];
        sums[2] = s1[0] / s2[0];   // ratio
    }
}

// ---------------------------------------------------------------------------
// Kernel 5: C = cost_kp + ratio * cost_desc   (in-place into cost_kp)
// ---------------------------------------------------------------------------
__global__ void fuse_kernel(const int* __restrict__ n_keep_p,
                            float* __restrict__ Cmat,
                            const float* __restrict__ cost_desc,
                            const float* __restrict__ sums) {
    const int n = *n_keep_p;
    const int total = n * n;
    const float ratio = sums[2];
    for (int idx = blockIdx.x * blockDim.x + threadIdx.x; idx < total;
         idx += gridDim.x * blockDim.x) {
        Cmat[idx] += ratio * cost_desc[idx];
    }
}

// ---------------------------------------------------------------------------
// Kernel 6: Hungarian (JV augmenting path, fp64 like reference), one block
// of 512 threads (16 wave32s); then final losses.
// ---------------------------------------------------------------------------
__global__ void hungarian_kernel(const int* __restrict__ n_keep_p,
                                 const float* __restrict__ Cmat,
                                 const float* __restrict__ f_pred,
                                 const float* __restrict__ f_cal,
                                 const int* __restrict__ keep_idx,
                                 const float* __restrict__ feat,
                                 const float* __restrict__ tf_feat,
                                 float* __restrict__ out) {
    const double INF = 1e18;
    const int n = *n_keep_p;
    const int m = n;
    const int T = 512;
    const int t = threadIdx.x;

    __shared__ double u[MAXN + 1], v[MAXN + 1], minv[MAXN + 1];
    __shared__ int p[MAXN + 1], way[MAXN + 1];
    __shared__ unsigned char used[MAXN + 1];
    __shared__ double red_val[512];
    __shared__ int red_idx[512];
    __shared__ int col4row[MAXN];
    __shared__ int s_j0;

    if (n > 0) {
        for (int j = t; j <= m; j += T) { u[j] = 0.0; v[j] = 0.0; p[j] = 0; way[j] = 0; }
        __syncthreads();

        for (int i = 1; i <= n; ++i) {
            if (t == 0) { p[0] = i; s_j0 = 0; }
            for (int j = t; j <= m; j += T) { minv[j] = INF; used[j] = 0; }
            __syncthreads();

            while (true) {
                const int j0 = s_j0;
                if (t == 0) used[j0] = 1;
                __syncthreads();
                const int i0 = p[j0];
                const double ui0 = u[i0];
                __builtin_prefetch(&Cmat[(i0 - 1) * n], 0, 0);

                // relax minv for unused columns
                for (int j = 1 + t; j <= m; j += T) {
                    if (!used[j]) {
                        const double cur =
                            (double)Cmat[(i0 - 1) * n + (j - 1)] - ui0 - v[j];
                        if (cur < minv[j]) { minv[j] = cur; way[j] = j0; }
                    }
                }
                __syncthreads();

                // argmin over j in [1,m], first-occurrence tie-break
                double bv = INF;
                int bi = m;
                for (int j = 1 + t; j <= m; j += T) {
                    const double val = used[j] ? INF : minv[j];
                    if (val < bv || (val == bv && j < bi)) { bv = val; bi = j; }
                }
                red_val[t] = bv; red_idx[t] = bi;
                __syncthreads();
                for (int s = 256; s > 0; s >>= 1) {
                    if (t < s) {
                        const double v2 = red_val[t + s];
                        const int i2 = red_idx[t + s];
                        if (v2 < red_val[t] ||
                            (v2 == red_val[t] && i2 < red_idx[t])) {
                            red_val[t] = v2; red_idx[t] = i2;
                        }
                    }
                    __syncthreads();
                }
                const int j1 = red_idx[0];
                const double delta = red_val[0];
                __syncthreads();

                // dual updates (p[j] distinct for used j -> race-free)
                for (int j = t; j <= m; j += T) {
                    if (used[j]) { u[p[j]] += delta; v[j] -= delta; }
                    else          { minv[j] -= delta; }
                }
                __syncthreads();
                if (t == 0) s_j0 = j1;
                __syncthreads();
                if (p[s_j0] == 0) break;
            }

            // augment along the alternating path
            if (t == 0) {
                int j0 = s_j0;
                while (j0 != 0) {
                    const int j1 = way[j0];
                    p[j0] = p[j1];
                    j0 = j1;
                }
            }
            __syncthreads();
        }
    }

    for (int j = 1 + t; j <= m; j += T) col4row[p[j] - 1] = j - 1;
    __syncthreads();

    // final deterministic sums
    double accC = 0.0, accKp = 0.0, accDesc = 0.0;
    for (int i = t; i < n; i += T) {
        const int cj = col4row[i];
        accC += (double)Cmat[i * n + cj];
        accKp += fabs((double)f_pred[i * 2 + 0] - (double)f_cal[cj * 2 + 0]) +
                 fabs((double)f_pred[i * 2 + 1] - (double)f_cal[cj * 2 + 1]);
    }
    for (int e = t; e < n * DESC; e += T) {
        const int i = e >> 8;
        const int d = e & (DESC - 1);
        const int cj = col4row[i];
        const int ra = keep_idx[i], rb = keep_idx[cj];
        accDesc += fabs((double)feat[ra * FEAT_STRIDE + 2 + d] -
                        (double)tf_feat[rb * FEAT_STRIDE + 2 + d]);
    }
    __shared__ double r0[512], r1[512], r2[512];
    r0[t] = accC; r1[t] = accKp; r2[t] = accDesc;
    __syncthreads();
    for (int s = 256; s > 0; s >>= 1) {
        if (t < s) { r0[t] += r0[t + s]; r1[t] += r1[t + s]; r2[t] += r2[t + s]; }
        __syncthreads();
    }
    if (t == 0) {
        const float inv = (n > 0) ? 1.0f / (float)n : 0.0f;
        out[0] = (float)r1[0] * inv;   // loss_kp
        out[1] = (float)r2[0] * inv;   // loss_desc
        out[2] = (float)r0[0];         // MyCost
    }
}

// ---------------------------------------------------------------------------
// Host-side launcher
// ---------------------------------------------------------------------------
extern "C" void kernel_launch(void* const* d_in, const int* in_sizes, int n_in,
                              void* d_out, int out_size, void* d_ws, size_t ws_size,
                              hipStream_t stream) {
    const float* theta   = (const float*)d_in[0];
    const float* feat    = (const float*)d_in[1];   // (B,K,258)
    const float* tf_feat = (const float*)d_in[2];   // (B,K,258)
    const int*   wptr    = (const int*)d_in[3];
    const int*   hptr    = (const int*)d_in[4];
    float* out = (float*)d_out;

    char* ws = (char*)d_ws;
    int*   n_keep    = (int*)(ws + 0);
    float* sums      = (float*)(ws + 16);                 // sum_kp, sum_desc, ratio
    int*   keep_idx  = (int*)(ws + 32);                   // 512 ints
    float* flat_cal  = (float*)(ws + 32 + 2048);          // 512x2
    float* f_pred    = (float*)(ws + 32 + 2048 + 4096);   // 512x2
    float* f_cal     = (float*)(ws + 32 + 2048 + 8192);   // 512x2
    float* cost_kp   = (float*)(ws + 16384);              // n*n (becomes C)
    float* cost_desc = (float*)(ws + 16384 + (size_t)MAXN * MAXN * 4);

    transform_kernel<<<1, 256, 0, stream>>>(theta, feat, wptr, hptr, flat_cal);
    compact_kernel<<<1, BK, 0, stream>>>(tf_feat, flat_cal, n_keep, keep_idx,
                                         f_pred, f_cal);
    cdist_kernel<<<dim3(MAXN / 16, MAXN / 16), dim3(16, 16), 0, stream>>>(
        feat, tf_feat, n_keep, keep_idx, f_pred, f_cal, cost_kp, cost_desc);
    reduce_kernel<<<1, 1024, 0, stream>>>(n_keep, cost_kp, cost_desc, sums);
    fuse_kernel<<<256, 256, 0, stream>>>(n_keep, cost_kp, cost_desc, sums);
    hungarian_kernel<<<1, 512, 0, stream>>>(n_keep, cost_kp, f_pred, f_cal,
                                            keep_idx, feat, tf_feat, out);
}